// TypedGATLayer_4303557230663
// MI455X (gfx1250) — compile-verified
//
#include <hip/hip_runtime.h>
#include <hip/hip_bf16.h>
#include <math.h>

#define DIM   128
#define HEADS 8
#define CH    16
#define NET   8
#define NEG_SLOPE 0.2f
#define LN_EPS    1e-5f

typedef float v2f __attribute__((ext_vector_type(2)));
typedef float v8f __attribute__((ext_vector_type(8)));

// Order-preserving encoding of f32 into u32 so atomicMax(uint) == float max.
__device__ __forceinline__ unsigned enc_f32(float f) {
    unsigned u = __float_as_uint(f);
    return (u & 0x80000000u) ? ~u : (u | 0x80000000u);
}
__device__ __forceinline__ float dec_f32(unsigned e) {
    unsigned u = (e & 0x80000000u) ? (e & 0x7FFFFFFFu) : ~e;
    return __uint_as_float(u);
}

__device__ __forceinline__ float lrelu(float v) {
    return v > 0.0f ? v : NEG_SLOPE * v;
}

// ---------------------------------------------------------------------------
// Kernel 1: fused xl = x@Wl + bl, xr = x@Wr + br via V_WMMA_F32_16X16X4_F32.
// Block = 256 threads = 8 waves. Block owns 16 rows of x (staged in LDS);
// wave w owns output cols [16w, 16w+16) and accumulates BOTH weight matrices.
// A-frag (16x4 f32): lane L<16 holds row L, {K, K+1}; lanes 16..31 hold {K+2,K+3}.
// B-frag (4x16 f32): lane holds col (lane&15), same K split by lane half.
// C/D (16x16 f32): VGPR r -> row r + 8*(lane>>4), col (lane&15).
// ---------------------------------------------------------------------------
__global__ __launch_bounds__(256) void k_linear_wmma(
    const float* __restrict__ x,
    const float* __restrict__ Wl, const float* __restrict__ bl,
    const float* __restrict__ Wr, const float* __restrict__ br,
    float* __restrict__ xl, float* __restrict__ xr, int n)
{
    __shared__ float xs[16][132];   // 132 pad -> stride 4 banks per row
    const int r0 = blockIdx.x * 16;
    const int t  = threadIdx.x;

    // Stage 16x128 tile of x into LDS (each thread moves 8 floats).
    {
        int row = t >> 4;
        int cb  = (t & 15) * 8;
        float4 a = make_float4(0.f, 0.f, 0.f, 0.f), b = a;
        if (r0 + row < n) {
            const float* p = x + (size_t)(r0 + row) * DIM + cb;
            a = *(const float4*)(p);
            b = *(const float4*)(p + 4);
        }
        *(float4*)&xs[row][cb]     = a;
        *(float4*)&xs[row][cb + 4] = b;
    }
    __syncthreads();

    const int wave = t >> 5;
    const int lane = t & 31;
    const int hh   = lane >> 4;       // which half of the wave
    const int lr   = lane & 15;
    const int n0   = wave * 16;       // this wave's output column tile

    v8f accL = {};
    v8f accR = {};
    for (int k = 0; k < DIM; k += 4) {
        const int kk = k + 2 * hh;
        v2f a;  a.x  = xs[lr][kk];
                a.y  = xs[lr][kk + 1];
        v2f bL; bL.x = Wl[(size_t)kk * DIM + n0 + lr];
                bL.y = Wl[(size_t)(kk + 1) * DIM + n0 + lr];
        v2f bR; bR.x = Wr[(size_t)kk * DIM + n0 + lr];
                bR.y = Wr[(size_t)(kk + 1) * DIM + n0 + lr];
        accL = __builtin_amdgcn_wmma_f32_16x16x4_f32(false, a, false, bL,
                                                     (short)0, accL, false, false);
        accR = __builtin_amdgcn_wmma_f32_16x16x4_f32(false, a, false, bR,
                                                     (short)0, accR, false, false);
    }

    const float biasL = bl[n0 + lr];
    const float biasR = br[n0 + lr];
#pragma unroll
    for (int r = 0; r < 8; ++r) {
        int row = r0 + r + 8 * hh;
        if (row < n) {
            xl[(size_t)row * DIM + n0 + lr] = accL[r] + biasL;
            xr[(size_t)row * DIM + n0 + lr] = accR[r] + biasR;
        }
    }
}

// ---------------------------------------------------------------------------
// Kernel 2: per-dst degree + edge_attr sums (L2-resident f32 atomics).
// ---------------------------------------------------------------------------
__global__ void k_edge_stats(const int* __restrict__ dst,
                             const float* __restrict__ edge_attr,
                             float* __restrict__ deg,
                             float* __restrict__ attr_sum, int E)
{
    int e = blockIdx.x * blockDim.x + threadIdx.x;
    if (e >= E) return;
    int d = dst[e];
    atomicAdd(&deg[d], 1.0f);
#pragma unroll
    for (int tt = 0; tt < NET; ++tt)
        atomicAdd(&attr_sum[(size_t)d * NET + tt], edge_attr[(size_t)e * NET + tt]);
}

// Kernel 3: self-loop attr = attr_sum / max(deg, 1)
__global__ void k_loop_attr(const float* __restrict__ deg,
                            const float* __restrict__ attr_sum,
                            float* __restrict__ loop_attr, int n)
{
    int i = blockIdx.x * blockDim.x + threadIdx.x;
    if (i >= n) return;
    float inv = 1.0f / fmaxf(deg[i], 1.0f);
#pragma unroll
    for (int tt = 0; tt < NET; ++tt)
        loop_attr[(size_t)i * NET + tt] = attr_sum[(size_t)i * NET + tt] * inv;
}

// ---------------------------------------------------------------------------
// Kernel 4: one wave per (edge or self-loop). Lane owns 4 channels (head =
// lane/4). score[h] = att_h . leaky(xl[src]+xr[dst]+edge_attr@We); atomicMax
// into per-(dst,head) running max (uint-encoded).
// ---------------------------------------------------------------------------
__global__ __launch_bounds__(256) void k_score(
    const int* __restrict__ src, const int* __restrict__ dst,
    const float* __restrict__ edge_attr, const float* __restrict__ loop_attr,
    const float* __restrict__ xl, const float* __restrict__ xr,
    const float* __restrict__ We, const float* __restrict__ att,
    float* __restrict__ score, unsigned* __restrict__ smax,
    int E, int n)
{
    int wid  = (blockIdx.x * blockDim.x + threadIdx.x) >> 5;
    int lane = threadIdx.x & 31;
    int ET   = E + n;
    if (wid >= ET) return;

    int s, d; const float* ea;
    if (wid < E) { s = src[wid]; d = dst[wid]; ea = edge_attr + (size_t)wid * NET; }
    else         { s = d = wid - E;            ea = loop_attr + (size_t)(wid - E) * NET; }

    const int c0 = lane * 4;
    float4 xlv = *(const float4*)(xl + (size_t)s * DIM + c0);
    float4 xrv = *(const float4*)(xr + (size_t)d * DIM + c0);

    float4 ee = make_float4(0.f, 0.f, 0.f, 0.f);
#pragma unroll
    for (int tt = 0; tt < NET; ++tt) {
        float c  = ea[tt];
        float4 w = *(const float4*)(We + (size_t)tt * DIM + c0);
        ee.x += c * w.x; ee.y += c * w.y; ee.z += c * w.z; ee.w += c * w.w;
    }

    const int h = lane >> 2;
    float4 av = *(const float4*)(att + h * CH + (lane & 3) * 4);
    float p = lrelu(xlv.x + xrv.x + ee.x) * av.x
            + lrelu(xlv.y + xrv.y + ee.y) * av.y
            + lrelu(xlv.z + xrv.z + ee.z) * av.z
            + lrelu(xlv.w + xrv.w + ee.w) * av.w;
    p += __shfl_xor(p, 1);
    p += __shfl_xor(p, 2);

    if ((lane & 3) == 0) {
        score[(size_t)wid * HEADS + h] = p;
        atomicMax(&smax[(size_t)d * HEADS + h], enc_f32(p));
    }
}

// Kernel 5: expd = exp(score - smax[dst]); denom[dst] += expd (in place).
__global__ void k_expsum(const int* __restrict__ dst,
                         const unsigned* __restrict__ smax,
                         float* __restrict__ score,   // in: score, out: expd
                         float* __restrict__ denom, int E, int n)
{
    int idx = blockIdx.x * blockDim.x + threadIdx.x;
    int ET  = E + n;
    if (idx >= ET * HEADS) return;
    int e = idx >> 3, h = idx & 7;
    int d = (e < E) ? dst[e] : (e - E);
    float sm = dec_f32(smax[(size_t)d * HEADS + h]);
    float ex = expf(score[idx] - sm);
    score[idx] = ex;
    atomicAdd(&denom[(size_t)d * HEADS + h], ex);
}

// Kernel 6: out[dst] += xl[src] * alpha. One wave per edge; all accumulators
// stay resident in the 192MB L2 so the 100M+ f32 atomics never hit HBM.
__global__ __launch_bounds__(256) void k_aggregate(
    const int* __restrict__ src, const int* __restrict__ dst,
    const float* __restrict__ xl, const float* __restrict__ expd,
    const float* __restrict__ denom, float* __restrict__ out, int E, int n)
{
    int wid  = (blockIdx.x * blockDim.x + threadIdx.x) >> 5;
    int lane = threadIdx.x & 31;
    int ET   = E + n;
    if (wid >= ET) return;
    int s, d;
    if (wid < E) { s = src[wid]; d = dst[wid]; } else { s = d = wid - E; }
    int h = lane >> 2;
    float alpha = expd[(size_t)wid * HEADS + h] / denom[(size_t)d * HEADS + h];
    int c0 = lane * 4;
    float4 xlv = *(const float4*)(xl + (size_t)s * DIM + c0);
    float* o = out + (size_t)d * DIM + c0;
    atomicAdd(o + 0, xlv.x * alpha);
    atomicAdd(o + 1, xlv.y * alpha);
    atomicAdd(o + 2, xlv.z * alpha);
    atomicAdd(o + 3, xlv.w * alpha);
}

// Kernel 7: + bias, LayerNorm over 128, ELU. One wave per node, in place.
__global__ __launch_bounds__(256) void k_finalize(
    float* __restrict__ out, const float* __restrict__ bias,
    const float* __restrict__ ln_w, const float* __restrict__ ln_b, int n)
{
    int wid  = (blockIdx.x * blockDim.x + threadIdx.x) >> 5;
    int lane = threadIdx.x & 31;
    if (wid >= n) return;
    int c0 = lane * 4;
    float* o = out + (size_t)wid * DIM + c0;
    float4 v  = *(float4*)o;
    float4 bv = *(const float4*)(bias + c0);
    v.x += bv.x; v.y += bv.y; v.z += bv.z; v.w += bv.w;

    float s  = v.x + v.y + v.z + v.w;
    float s2 = v.x * v.x + v.y * v.y + v.z * v.z + v.w * v.w;
#pragma unroll
    for (int m = 1; m < 32; m <<= 1) {
        s  += __shfl_xor(s,  m);
        s2 += __shfl_xor(s2, m);
    }
    float mu  = s * (1.0f / DIM);
    float var = s2 * (1.0f / DIM) - mu * mu;
    float rs  = rsqrtf(var + LN_EPS);

    float4 wv = *(const float4*)(ln_w + c0);
    float4 lb = *(const float4*)(ln_b + c0);
    float t0 = (v.x - mu) * rs * wv.x + lb.x;
    float t1 = (v.y - mu) * rs * wv.y + lb.y;
    float t2 = (v.z - mu) * rs * wv.z + lb.z;
    float t3 = (v.w - mu) * rs * wv.w + lb.w;
    v.x = t0 > 0.f ? t0 : expm1f(t0);
    v.y = t1 > 0.f ? t1 : expm1f(t1);
    v.z = t2 > 0.f ? t2 : expm1f(t2);
    v.w = t3 > 0.f ? t3 : expm1f(t3);
    *(float4*)o = v;
}

// ---------------------------------------------------------------------------
extern "C" void kernel_launch(void* const* d_in, const int* in_sizes, int n_in,
                              void* d_out, int out_size, void* d_ws, size_t ws_size,
                              hipStream_t stream)
{
    (void)n_in; (void)out_size; (void)ws_size;
    const float* x         = (const float*)d_in[0];
    const int*   src       = (const int*)  d_in[1];
    const int*   dst       = (const int*)  d_in[2];
    const float* edge_attr = (const float*)d_in[3];
    const float* Wl        = (const float*)d_in[4];
    const float* blv       = (const float*)d_in[5];
    const float* Wr        = (const float*)d_in[6];
    const float* brv       = (const float*)d_in[7];
    const float* We        = (const float*)d_in[8];
    const float* att       = (const float*)d_in[9];
    const float* bias      = (const float*)d_in[10];
    const float* ln_w      = (const float*)d_in[11];
    const float* ln_b      = (const float*)d_in[12];

    const int N  = in_sizes[0] / DIM;
    const int E  = in_sizes[1];
    const int ET = E + N;

    // Workspace layout (floats); all region sizes kept 16B-aligned.
    float* ws = (float*)d_ws;
    size_t off = 0;
    float*    deg       = ws + off; off += (size_t)N;            off = (off + 3) & ~(size_t)3;
    float*    attr_sum  = ws + off; off += (size_t)N * NET;
    float*    loop_attr = ws + off; off += (size_t)N * NET;
    unsigned* smax      = (unsigned*)(ws + off); off += (size_t)N * NET;
    float*    denom     = ws + off; off += (size_t)N * NET;
    size_t zero_floats  = off;                       // deg .. denom (contiguous)
    float*    xl        = ws + off; off += (size_t)N * DIM;
    float*    xr        = ws + off; off += (size_t)N * DIM;
    float*    score     = ws + off; off += (size_t)ET * HEADS;   // reused as expd

    float* out = (float*)d_out;

    // Re-zero accumulators every call (graph replays must be deterministic).
    hipMemsetAsync(ws,  0, zero_floats * sizeof(float), stream);     // smax: 0u == -inf sentinel
    hipMemsetAsync(out, 0, (size_t)N * DIM * sizeof(float), stream);

    k_linear_wmma<<<(N + 15) / 16, 256, 0, stream>>>(x, Wl, blv, Wr, brv, xl, xr, N);
    k_edge_stats <<<(E + 255) / 256, 256, 0, stream>>>(dst, edge_attr, deg, attr_sum, E);
    k_loop_attr  <<<(N + 255) / 256, 256, 0, stream>>>(deg, attr_sum, loop_attr, N);
    k_score      <<<((size_t)ET * 32 + 255) / 256, 256, 0, stream>>>(
        src, dst, edge_attr, loop_attr, xl, xr, We, att, score, smax, E, N);
    k_expsum     <<<((size_t)ET * HEADS + 255) / 256, 256, 0, stream>>>(
        dst, smax, score, denom, E, N);
    k_aggregate  <<<((size_t)ET * 32 + 255) / 256, 256, 0, stream>>>(
        src, dst, xl, score, denom, out, E, N);
    k_finalize   <<<((size_t)N * 32 + 255) / 256, 256, 0, stream>>>(
        out, bias, ln_w, ln_b, N);
}